// SNN_6184752906852
// MI455X (gfx1250) — compile-verified
//
#include <hip/hip_runtime.h>

// SNN single-step: X = (V >= 1 + 1.8a); current = sum_d Xd[d]@W_int[d] + FF@W_ff;
// V_new = 0.9*V*(1-X) + current. Out = [2,B,N] (X then V_new).
// B=16, N=4096, N_FF=2048, D=8.  f32 WMMA 16x16x4, HBM-bound (~537MB weights).
// B-loads use immediate offsets (24-bit IOFFSET) from one base pointer per wave
// to minimize VALU address math; W stream is non-temporal to preserve Xd/FF in L2.

typedef __attribute__((ext_vector_type(2))) float v2f;
typedef __attribute__((ext_vector_type(8))) float v8f;

namespace {
constexpr int   kB    = 16;
constexpr int   kN    = 4096;
constexpr int   kNFF  = 2048;
constexpr float kALPHA = 0.9f;
constexpr float kBETA  = 1.8f;
constexpr float kTH0   = 1.0f;

constexpr int kWaves   = 18;            // 8 delays x 2 K-halves + 2 FF K-halves
constexpr int kThreads = kWaves * 32;   // wave32
constexpr int kUnroll  = 8;             // k-steps (of 4) per unrolled body
}

__global__ __launch_bounds__(kThreads)
void snn_fused_wmma(const float* __restrict__ V,
                    const float* __restrict__ a,
                    const float* __restrict__ Xd,    // [D,B,N]
                    const float* __restrict__ FF,    // [B,N_FF]
                    const float* __restrict__ Wint,  // [D,N,N]
                    const float* __restrict__ Wff,   // [N_FF,N]
                    float* __restrict__ out)         // [2,B,N]
{
    __shared__ float red[kWaves * 256];

    const int tid  = threadIdx.x;
    const int wave = tid >> 5;
    const int lane = tid & 31;
    const int half = lane >> 4;     // 0 or 1
    const int idx  = lane & 15;     // M for A rows, N for B/C/D cols
    const int n0   = blockIdx.x * 16;

    v8f acc = {};   // 16x16 f32 accumulator tile (8 VGPRs)

    if (wave < 16) {
        // Internal synapse GEMM slice: delay d, one K-half of 2048
        const int d     = wave >> 1;
        const int kbase = (wave & 1) * 2048;
        // Bases already folded with half/idx so unrolled offsets are constants.
        const float* __restrict__ Ap =
            Xd + (size_t)d * kB * kN + (size_t)idx * kN + (kbase + 2 * half);
        const float* __restrict__ Bp =
            Wint + (size_t)d * kN * kN + (size_t)(kbase + 2 * half) * kN + (n0 + idx);
        for (int it = 0; it < 2048 / (4 * kUnroll); ++it) {
            #pragma unroll
            for (int u = 0; u < kUnroll; ++u) {
                // A: {Xd[..][k+2h], Xd[..][k+2h+1]} -> global_load_b64, imm offset 16*u
                v2f av = *(const v2f*)(Ap + 4 * u);
                // B: rows k+2h and k+2h+1, col n0+idx -> imm offsets u*64KB (+16KB)
                v2f bv;
                bv.x = __builtin_nontemporal_load(Bp + (size_t)(4 * u)     * kN);
                bv.y = __builtin_nontemporal_load(Bp + (size_t)(4 * u + 1) * kN);
                acc = __builtin_amdgcn_wmma_f32_16x16x4_f32(
                    false, av, false, bv, (short)0, acc, false, false);
            }
            Ap += 4 * kUnroll;
            Bp += (size_t)(4 * kUnroll) * kN;
        }
    } else {
        // Feedforward GEMM slice: one K-half of 1024 over N_FF=2048
        const int kbase = (wave - 16) * 1024;
        const float* __restrict__ Ap = FF + (size_t)idx * kNFF + (kbase + 2 * half);
        const float* __restrict__ Bp =
            Wff + (size_t)(kbase + 2 * half) * kN + (n0 + idx);
        for (int it = 0; it < 1024 / (4 * kUnroll); ++it) {
            #pragma unroll
            for (int u = 0; u < kUnroll; ++u) {
                v2f av = *(const v2f*)(Ap + 4 * u);
                v2f bv;
                bv.x = __builtin_nontemporal_load(Bp + (size_t)(4 * u)     * kN);
                bv.y = __builtin_nontemporal_load(Bp + (size_t)(4 * u + 1) * kN);
                acc = __builtin_amdgcn_wmma_f32_16x16x4_f32(
                    false, av, false, bv, (short)0, acc, false, false);
            }
            Ap += 4 * kUnroll;
            Bp += (size_t)(4 * kUnroll) * kN;
        }
    }

    // Spill partial tile: C/D layout is (M = j + 8*half, N = idx) for VGPR j
    #pragma unroll
    for (int j = 0; j < 8; ++j) {
        red[wave * 256 + (j + 8 * half) * 16 + idx] = acc[j];
    }
    __syncthreads();

    // Reduce 18 partial tiles + fused ALIF membrane update (one thread/element)
    if (tid < 256) {
        float cur = 0.0f;
        #pragma unroll
        for (int w = 0; w < kWaves; ++w) cur += red[w * 256 + tid];

        const int m = tid >> 4;             // batch row
        const int n = tid & 15;             // col in tile
        const size_t off = (size_t)m * kN + (size_t)(n0 + n);

        const float v  = V[off];
        const float ad = a[off];
        const float theta = kTH0 + kBETA * ad;
        const float X = (v >= theta) ? 1.0f : 0.0f;
        const float vnew = kALPHA * v * (1.0f - X) + cur;

        out[off]                   = X;     // out[0] : spikes
        out[(size_t)kB * kN + off] = vnew;  // out[1] : membrane
    }
}

extern "C" void kernel_launch(void* const* d_in, const int* in_sizes, int n_in,
                              void* d_out, int out_size, void* d_ws, size_t ws_size,
                              hipStream_t stream) {
    (void)in_sizes; (void)n_in; (void)out_size; (void)d_ws; (void)ws_size;
    const float* V    = (const float*)d_in[0];
    const float* a    = (const float*)d_in[1];
    const float* Xd   = (const float*)d_in[2];
    const float* FF   = (const float*)d_in[3];
    const float* Wint = (const float*)d_in[4];
    const float* Wff  = (const float*)d_in[5];
    float* out = (float*)d_out;

    dim3 grid(kN / 16);          // 256 output tiles of 16x16
    dim3 block(kThreads);        // 18 waves
    snn_fused_wmma<<<grid, block, 0, stream>>>(V, a, Xd, FF, Wint, Wff, out);
}